// HEAQNetwork_77687368450518
// MI455X (gfx1250) — compile-verified
//
#include <hip/hip_runtime.h>
#include <math.h>

// ---------------------------------------------------------------------------
// 4-qubit VQC batch simulator for MI455X (gfx1250, wave32).
//
// Per layer:  state' = U_l * (RX(theta_0..3) state)
//   - RX part is per-element -> VALU (+ ds_swizzle for the MSB qubit)
//   - U_l = CNOTchain * prod RY(weights[l,q]) is a fixed real 16x16 matrix
//     -> applied with chained V_WMMA_F32_16X16X4_F32 (K=4 x 4 = 16),
//        once for real parts, once for imag parts, 16 batch elems per wave.
// State layout == WMMA C/D f32 16x16 layout:
//   lane n   (n<16): amplitudes M=0..7  of element n  in VGPRs 0..7
//   lane n+16      : amplitudes M=8..15 of element n  in VGPRs 0..7
// so the WMMA result needs no layout conversion between layers.
//
// Trig: cos/sin(atan(u)/2) computed in closed form:
//   r = 1/sqrt(1+u^2);  c = sqrt((1+r)/2);  s = copysign(sqrt((1-r)/2), u)
// -> no libm atan/sincos expansion, just v_rsq/v_sqrt.
// ---------------------------------------------------------------------------

typedef float v2f __attribute__((ext_vector_type(2)));
typedef float v4f __attribute__((ext_vector_type(4)));
typedef float v8f __attribute__((ext_vector_type(8)));

#define NQ 4
#define NL 3

// lanes 0..15 <-> 16..31 (group-of-32 swizzle, xor=0x10, and=0x1f)
__device__ __forceinline__ float swapx16(float v) {
  int r = __builtin_amdgcn_ds_swizzle(__builtin_bit_cast(int, v), 0x401f);
  return __builtin_bit_cast(float, r);
}

// ---------------------------------------------------------------------------
// Setup: build U_l[i][j] = < i | CNOT(2,3)CNOT(1,2)CNOT(0,1) RY3 RY2 RY1 RY0 | j >
// (real 16x16, one per layer), stored row-major at U + l*256.
// Qubit q is bit (3-q) of the amplitude index (reference moveaxis convention).
// ---------------------------------------------------------------------------
__global__ void heaq_build_unitaries(const float* __restrict__ weights,
                                     float* __restrict__ U) {
  int t = threadIdx.x;
  if (t >= NL * 16) return;
  int l = t >> 4;
  int j = t & 15;
  float col[16];
#pragma unroll
  for (int i = 0; i < 16; ++i) col[i] = 0.0f;
  col[j] = 1.0f;
  // RY on qubits 0..3:  out0 = c*a0 - s*a1 ; out1 = s*a0 + c*a1
#pragma unroll
  for (int q = 0; q < NQ; ++q) {
    float h = 0.5f * weights[l * NQ + q];
    float c = cosf(h), s = sinf(h);
    int m = 8 >> q;
#pragma unroll
    for (int p = 0; p < 16; ++p) {
      if (p & m) continue;
      float a0 = col[p], a1 = col[p | m];
      col[p]     = c * a0 - s * a1;
      col[p | m] = s * a0 + c * a1;
    }
  }
  // CNOT(ctrl, ctrl+1): swap target-bit pair where ctrl bit is set
#pragma unroll
  for (int ctrl = 0; ctrl < NQ - 1; ++ctrl) {
    int cm = 8 >> ctrl, tm = 8 >> (ctrl + 1);
#pragma unroll
    for (int p = 0; p < 16; ++p) {
      if ((p & cm) && !(p & tm)) {
        float tv = col[p]; col[p] = col[p | tm]; col[p | tm] = tv;
      }
    }
  }
#pragma unroll
  for (int i = 0; i < 16; ++i) U[l * 256 + i * 16 + j] = col[i];
}

// In-lane RX gate on qubit with amplitude mask m (4, 2 or 1):
//   out_r = c*a_r + s*partner_i ; out_i = c*a_i - s*partner_r
__device__ __forceinline__ void rx_inlane(float (&sr)[8], float (&si)[8],
                                          float c, float s, int m) {
#pragma unroll
  for (int p = 0; p < 8; ++p) {
    if (p & m) continue;
    int q = p | m;
    float a0r = sr[p], a0i = si[p], a1r = sr[q], a1i = si[q];
    sr[p] = fmaf(c, a0r,  s * a1i);
    si[p] = fmaf(c, a0i, -s * a1r);
    sr[q] = fmaf(c, a1r,  s * a0i);
    si[q] = fmaf(c, a1i, -s * a0r);
  }
}

// Build B-fragments (4x16 K-chunks of the 16x16 state-in-columns matrix).
// B VGPR0: lanes0-15 = row K0, lanes16-31 = row K0+2 ; VGPR1: K0+1 / K0+3.
__device__ __forceinline__ void build_bfrags(const float (&st)[8], bool hi,
                                             v2f (&Bf)[4]) {
  float w0 = swapx16(st[0]), w1 = swapx16(st[1]);
  float w2 = swapx16(st[2]), w3 = swapx16(st[3]);
  float w4 = swapx16(st[4]), w5 = swapx16(st[5]);
  float w6 = swapx16(st[6]), w7 = swapx16(st[7]);
  Bf[0][0] = hi ? w2 : st[0];  Bf[0][1] = hi ? w3 : st[1];   // K = 0..3
  Bf[1][0] = hi ? w6 : st[4];  Bf[1][1] = hi ? w7 : st[5];   // K = 4..7
  Bf[2][0] = hi ? st[2] : w0;  Bf[2][1] = hi ? st[3] : w1;   // K = 8..11
  Bf[3][0] = hi ? st[6] : w4;  Bf[3][1] = hi ? st[7] : w5;   // K = 12..15
}

__global__ void __launch_bounds__(256)
heaq_qsim_kernel(const float* __restrict__ x,
                 const float* __restrict__ w_input,
                 const float* __restrict__ w_output,
                 const float* __restrict__ U,
                 float* __restrict__ out, int B) {
  const int lane = threadIdx.x & 31;
  const int wave = (int)((blockIdx.x * blockDim.x + threadIdx.x) >> 5);
  const long base = (long)wave * 16;
  if (base >= B) return;                 // whole-wave uniform exit (EXEC stays full)
  const bool hi = lane >= 16;
  const int  col = lane & 15;
  const long e  = base + col;
  const long ec = (e < B) ? e : (long)(B - 1);   // clamp loads, keep EXEC full

  // --- per-element RX half-angle cos/sin, closed form (no atan/sincos) ---
  const v4f xv = *reinterpret_cast<const v4f*>(x + ec * 4);
  const v4f wi = *reinterpret_cast<const v4f*>(w_input);
  float cg[4], sg[4];
#pragma unroll
  for (int q = 0; q < 4; ++q) {
    float u = xv[q] * wi[q];
    float r = __builtin_amdgcn_rsqf(fmaf(u, u, 1.0f));   // cos(atan u)
    float c = __builtin_amdgcn_sqrtf(0.5f * (1.0f + r));
    float s = __builtin_amdgcn_sqrtf(0.5f * (1.0f - r));
    cg[q] = c;
    sg[q] = __builtin_copysignf(s, u);
  }

  // --- A-fragments of U_l (16x4 chunks), uniform across waves -> cache hits ---
  // lane L<16: row L, cols 4kk+{0,1}; lane L>=16: row L-16, cols 4kk+{2,3}
  v2f Af[NL][4];
#pragma unroll
  for (int l = 0; l < NL; ++l)
#pragma unroll
    for (int kk = 0; kk < 4; ++kk)
      Af[l][kk] = *reinterpret_cast<const v2f*>(
          U + l * 256 + col * 16 + 4 * kk + (hi ? 2 : 0));

  // --- state |0000>: amplitude M=0 (VGPR0, low half) = 1 ---
  float sr[8], si[8];
#pragma unroll
  for (int r = 0; r < 8; ++r) { sr[r] = 0.0f; si[r] = 0.0f; }
  sr[0] = hi ? 0.0f : 1.0f;

#pragma unroll
  for (int l = 0; l < NL; ++l) {
    // RX on qubit 0 (mask 8): partner lives in the other lane half
#pragma unroll
    for (int r = 0; r < 8; ++r) {
      float pr = swapx16(sr[r]);
      float pi = swapx16(si[r]);
      sr[r] = fmaf(cg[0], sr[r],  sg[0] * pi);
      si[r] = fmaf(cg[0], si[r], -sg[0] * pr);
    }
    // RX on qubits 1..3: in-lane register pairs
    rx_inlane(sr, si, cg[1], sg[1], 4);
    rx_inlane(sr, si, cg[2], sg[2], 2);
    rx_inlane(sr, si, cg[3], sg[3], 1);

    // Apply U_l with WMMA:  D = sum_kk A(kk) x B(kk), real & imag separately
    v2f Br[4], Bi[4];
    build_bfrags(sr, hi, Br);
    build_bfrags(si, hi, Bi);
    v8f dr = {}, di = {};
#pragma unroll
    for (int kk = 0; kk < 4; ++kk) {
      dr = __builtin_amdgcn_wmma_f32_16x16x4_f32(false, Af[l][kk], false,
                                                 Br[kk], (short)0, dr,
                                                 false, false);
      di = __builtin_amdgcn_wmma_f32_16x16x4_f32(false, Af[l][kk], false,
                                                 Bi[kk], (short)0, di,
                                                 false, false);
    }
    // D layout == state layout: just unpack
#pragma unroll
    for (int r = 0; r < 8; ++r) { sr[r] = dr[r]; si[r] = di[r]; }
  }

  // --- probabilities and <Z0>, <Z1> ---
  float p[8];
#pragma unroll
  for (int r = 0; r < 8; ++r) p[r] = fmaf(sr[r], sr[r], si[r] * si[r]);
  float lo4 = (p[0] + p[1]) + (p[2] + p[3]);
  float hi4 = (p[4] + p[5]) + (p[6] + p[7]);
  float t = lo4 + hi4;            // sum over this half's 8 amplitudes
  float v = lo4 - hi4;            // signed by bit2 of amplitude index
  float z0 = t - swapx16(t);      // valid at lanes 0..15 (bit3 == lane half)
  float z1 = v + swapx16(v);      // valid at lanes 0..15

  if (!hi && e < B) {
    v2f o;
    o[0] = (1.0f + z0) * 0.5f * w_output[0];
    o[1] = (1.0f + z1) * 0.5f * w_output[1];
    *reinterpret_cast<v2f*>(out + 2 * e) = o;
  }
}

extern "C" void kernel_launch(void* const* d_in, const int* in_sizes, int n_in,
                              void* d_out, int out_size, void* d_ws, size_t ws_size,
                              hipStream_t stream) {
  const float* x        = (const float*)d_in[0];   // (B, 4)
  const float* w_input  = (const float*)d_in[1];   // (4,)
  const float* weights  = (const float*)d_in[2];   // (3, 4)
  const float* w_output = (const float*)d_in[3];   // (2,)
  float* out = (float*)d_out;                      // (B, 2)
  float* U   = (float*)d_ws;                       // 3 * 16 * 16 floats

  int B = in_sizes[0] / 4;

  heaq_build_unitaries<<<1, 64, 0, stream>>>(weights, U);

  long waves   = ((long)B + 15) / 16;
  long threads = waves * 32;
  int  block   = 256;
  long grid    = (threads + block - 1) / block;
  heaq_qsim_kernel<<<dim3((unsigned)grid), dim3(block), 0, stream>>>(
      x, w_input, w_output, U, out, B);
}